// Prediction_47949014892993
// MI455X (gfx1250) — compile-verified
//
#include <hip/hip_runtime.h>
#include <math.h>

// ---------------- problem geometry ----------------
#define NTOT   108864              // (96*96 + 48*48 + 24*24) * 3 anchors * 3 classes
#define NBATCH 32
#define BOXN   36288               // NTOT / 3  (boxes shared across the class dim)

// ---------------- radix sort config: 45-bit key, 5 passes of 9 bits -------
#define RBITS  9
#define RADIX  512                 // 1 << RBITS
#define TPB    512                 // 16 wave32's
#define EPT    4                   // elements per thread in scatter
#define TILE   2048                // TPB * EPT
#define TILES  54                  // ceil(NTOT / TILE)
#define NWAVE  16

#define SCORE_MIN_BITS 0x3D4CCCCDu   // __float_as_uint(0.05f)
#define REL_MAX        0x3FFFFFFu    // 26-bit field

typedef unsigned long long u64;
typedef unsigned int       u32;

__device__ __forceinline__ float sigf(float x) { return 1.0f / (1.0f + expf(-x)); }

// wave32 ballot (gfx1250 is wave32-only)
__device__ __forceinline__ unsigned wave_ballot(bool p) {
#if __has_builtin(__builtin_amdgcn_ballot_w32)
  return __builtin_amdgcn_ballot_w32(p);
#else
  return (unsigned)__ballot((int)p);
#endif
}

__device__ __forceinline__ unsigned lds_byte_off(void* p) {
  return (unsigned)(unsigned long long)((__attribute__((address_space(3))) void*)p);
}

// ---------------- kernel 1: decode (one thread per box = cell*anchor) -----
// key = [group(2b)=class+1|0  |  rel(26b) descending-score  |  elemIdx(17b)]
// Sorting keys ascending per batch yields the reference's final order directly.
__global__ __launch_bounds__(256) void k_decode(
    const float* __restrict__ o1, const float* __restrict__ a1,
    const float* __restrict__ f1, const float* __restrict__ s1,
    const float* __restrict__ o2, const float* __restrict__ a2,
    const float* __restrict__ f2, const float* __restrict__ s2,
    const float* __restrict__ o3, const float* __restrict__ a3,
    const float* __restrict__ f3, const float* __restrict__ s3,
    u64* __restrict__ keys, float4* __restrict__ boxes)
{
  long long tid = (long long)blockIdx.x * blockDim.x + threadIdx.x;
  if (tid >= (long long)NBATCH * BOXN) return;
  int b   = (int)(tid / BOXN);
  int box = (int)(tid % BOXN);

  int bl, cells; const float *op, *an, *of, *st;
  if (box < 27648)      { bl = box;         cells = 9216; op = o1; an = a1; of = f1; st = s1; }
  else if (box < 34560) { bl = box - 27648; cells = 2304; op = o2; an = a2; of = f2; st = s2; }
  else                  { bl = box - 34560; cells = 576;  op = o3; an = a3; of = f3; st = s3; }
  int cell = bl / 3, a = bl % 3;

  // channels [a*8 .. a*8+7] of this cell: (x,y,w,h,obj,c0,c1,c2) — 16B aligned
  const float4* o4 = (const float4*)(op + ((size_t)b * cells + cell) * 24 + a * 8);
  float4 lo = o4[0];                 // x, y, w, h
  float4 hi = o4[1];                 // obj, c0, c1, c2

  float sx = st[0], sy = st[1];
  float px = (sigf(lo.x) + of[cell * 2 + 0]) * sx;
  float py = (sigf(lo.y) + of[cell * 2 + 1]) * sy;
  float w  = expf(lo.z) * an[a * 2 + 0];
  float h  = expf(lo.w) * an[a * 2 + 1];
  boxes[(size_t)b * BOXN + box] =
      make_float4(px - 0.5f * w, py - 0.5f * h, px + 0.5f * w, py + 0.5f * h);

  float obj = sigf(hi.x);
  float scv[3] = { obj * sigf(hi.y), obj * sigf(hi.z), obj * sigf(hi.w) };

  u64* kp = keys + (size_t)b * NTOT + (size_t)box * 3;
  #pragma unroll
  for (int c = 0; c < 3; ++c) {
    u32 group = 0u, rel = 0u;
    float score = scv[c];
    if (score > 0.05f) {             // EXCEPT_THRESH; score in (0.05, 1.0]
      group = (u32)c + 1u;
      rel = REL_MAX - (__float_as_uint(score) - SCORE_MIN_BITS);  // fits 26 bits
    }
    kp[c] = ((u64)group << 43) | ((u64)rel << 17) | (u64)(u32)(box * 3 + c);
  }
}

// ---------------- radix pass kernel A: per-tile histogram ----------------
__global__ __launch_bounds__(TPB) void k_hist(const u64* __restrict__ src,
                                              u32* __restrict__ cnt, int shift)
{
  __shared__ u32 h[RADIX];
  int t = blockIdx.x, b = blockIdx.y, tid = threadIdx.x;
  h[tid] = 0u;
  __syncthreads();
  #pragma unroll
  for (int k = 0; k < EPT; ++k) {
    int e = t * TILE + k * TPB + tid;
    if (e < NTOT) {
      u32 dig = (u32)(src[(size_t)b * NTOT + e] >> shift) & (RADIX - 1u);
      atomicAdd(&h[dig], 1u);                                  // ds_add
    }
  }
  __syncthreads();
  cnt[((size_t)b * TILES + t) * RADIX + tid] = h[tid];
}

// ---------------- radix pass kernel B: per-batch digit/tile scan ----------
__global__ __launch_bounds__(RADIX) void k_scan(const u32* __restrict__ cnt,
                                                u32* __restrict__ off)
{
  __shared__ u32 tot[RADIX];
  __shared__ u32 base[RADIX];
  int b = blockIdx.x, d = threadIdx.x;
  u32 run = 0u;
  for (int t = 0; t < TILES; ++t) {
    __builtin_prefetch(&cnt[((size_t)b * TILES + t + 2) * RADIX + d], 0, 1);  // global_prefetch_b8
    u32 c = cnt[((size_t)b * TILES + t) * RADIX + d];
    off[((size_t)b * TILES + t) * RADIX + d] = run;
    run += c;
  }
  tot[d] = run;
  __syncthreads();
  if (d == 0) { u32 r = 0u; for (int i = 0; i < RADIX; ++i) { base[i] = r; r += tot[i]; } }
  __syncthreads();
  u32 add = base[d];
  for (int t = 0; t < TILES; ++t)
    off[((size_t)b * TILES + t) * RADIX + d] += add;
}

// ---------------- radix pass kernel C: stable scatter, 4 elems/thread -----
// Element order inside a tile is wave-major contiguous: wave w owns elements
// [w*128, w*128+128), processed as 4 rounds of 32 lanes -> (wave, round, lane)
// order equals memory order, so ranks are stable.
__global__ __launch_bounds__(TPB) void k_scatter(const u64* __restrict__ src,
                                                 u64* __restrict__ dst,
                                                 const u32* __restrict__ off,
                                                 int shift)
{
  __shared__ unsigned short wcnt[NWAVE][RADIX];   // per-wave running digit counts
  __shared__ u32            woff[NWAVE][RADIX];
  __shared__ u64            tile[TILE];
  int t = blockIdx.x, b = blockIdx.y, tid = threadIdx.x;
  int lane = tid & 31, wave = tid >> 5;

  { unsigned short* wz = &wcnt[0][0];
    #pragma unroll
    for (int i = tid; i < NWAVE * RADIX; i += TPB) wz[i] = 0; }

  // Stage the tile into LDS via the CDNA5 async copy path (ASYNCcnt-tracked).
  #pragma unroll
  for (int r = 0; r < EPT; ++r) {
    int li = wave * (32 * EPT) + r * 32 + lane;
    int e  = t * TILE + li;
    if (e < NTOT) {
      unsigned loff = lds_byte_off((void*)&tile[li]);
      u64 gaddr = (u64)(size_t)(src + (size_t)b * NTOT + e);
      asm volatile("global_load_async_to_lds_b64 %0, %1, off"
                   :: "v"(loff), "v"(gaddr) : "memory");
    }
  }
  asm volatile("s_wait_asynccnt 0" ::: "memory");
  __syncthreads();

  u64 kv[EPT]; u32 dg[EPT]; u32 rk[EPT]; bool av[EPT];
  const unsigned ltmask = (1u << lane) - 1u;
  #pragma unroll
  for (int r = 0; r < EPT; ++r) {
    int li = wave * (32 * EPT) + r * 32 + lane;
    int e  = t * TILE + li;
    bool act = e < NTOT;
    u64 key = act ? tile[li] : 0ull;
    u32 dig = (u32)(key >> shift) & (RADIX - 1u);

    unsigned m = wave_ballot(act);                 // stable in-wave rank, 9-bit digit
    #pragma unroll
    for (int bit = 0; bit < RBITS; ++bit) {
      unsigned bb = wave_ballot(((dig >> bit) & 1u) != 0u);
      m &= ((dig >> bit) & 1u) ? bb : ~bb;
    }
    u32 below = __popc(m & ltmask);
    u32 cbase = (u32)wcnt[wave][dig];              // ds_load precedes ds_store (in order)
    rk[r] = cbase + below;
    if (act && below == 0u)                        // digit-group leader bumps the count
      wcnt[wave][dig] = (unsigned short)(cbase + __popc(m));
    kv[r] = key; dg[r] = dig; av[r] = act;
  }
  __syncthreads();

  if (tid < RADIX) {
    u32 run = off[((size_t)b * TILES + t) * RADIX + tid];
    #pragma unroll
    for (int w = 0; w < NWAVE; ++w) { u32 c = wcnt[w][tid]; woff[w][tid] = run; run += c; }
  }
  __syncthreads();

  #pragma unroll
  for (int r = 0; r < EPT; ++r)
    if (av[r]) dst[(size_t)b * NTOT + woff[wave][dg[r]] + rk[r]] = kv[r];
}

// ---------------- kernel 5: suppression + clamping + output ----------------
__global__ __launch_bounds__(256) void k_output(const u64* __restrict__ keys,
                                                const float4* __restrict__ boxes,
                                                float* __restrict__ out)
{
  long long tid = (long long)blockIdx.x * blockDim.x + threadIdx.x;
  if (tid >= (long long)NBATCH * NTOT) return;
  int b = (int)(tid / NTOT), p = (int)(tid % NTOT);

  u64 k = keys[(size_t)b * NTOT + p];
  u32 group = (u32)(k >> 43);
  u32 idx   = (u32)k & 0x1FFFFu;
  float4 bx = boxes[(size_t)b * BOXN + idx / 3u];

  float id, sc; float4 ob;
  if (group == 0u) {                       // invalid item: raw box, -1 id/score
    id = -1.0f; sc = -1.0f; ob = bx;
  } else {
    u32 rel = (u32)(k >> 17) & REL_MAX;    // recover exact score bits
    float score = __uint_as_float((REL_MAX - rel) + SCORE_MIN_BITS);

    bool kept = true;
    if (p > 0) {
      u64 kp = keys[(size_t)b * NTOT + p - 1];
      if ((u32)(kp >> 43) == group) {      // adjacent pair within same class segment
        u32 pidx = (u32)kp & 0x1FFFFu;
        float4 pb = boxes[(size_t)b * BOXN + pidx / 3u];
        float xx1 = fmaxf(pb.x, bx.x), yy1 = fmaxf(pb.y, bx.y);
        float xx2 = fminf(pb.z, bx.z), yy2 = fminf(pb.w, bx.w);
        float inter = (xx2 - xx1 + 1.0f) * (yy2 - yy1 + 1.0f);   // no clamp (match ref)
        float a0 = (pb.z - pb.x + 1.0f) * (pb.w - pb.y + 1.0f);
        float a1 = (bx.z - bx.x + 1.0f) * (bx.w - bx.y + 1.0f);
        float ov = inter / (a0 + a1 - inter);
        kept = !(ov > 0.5f);               // NMS_THRESH
      }
    }
    if (kept) {                            // clamp(box * +1): negatives -> -1
      id = (float)(group - 1u); sc = score;
      ob.x = bx.x < 0.0f ? -1.0f : bx.x;
      ob.y = bx.y < 0.0f ? -1.0f : bx.y;
      ob.z = bx.z < 0.0f ? -1.0f : bx.z;
      ob.w = bx.w < 0.0f ? -1.0f : bx.w;
    } else {                               // clamp(box * -1)
      id = -1.0f; sc = -1.0f;
      ob.x = (-bx.x) < 0.0f ? -1.0f : -bx.x;
      ob.y = (-bx.y) < 0.0f ? -1.0f : -bx.y;
      ob.z = (-bx.z) < 0.0f ? -1.0f : -bx.z;
      ob.w = (-bx.w) < 0.0f ? -1.0f : -bx.w;
    }
  }
  size_t BN = (size_t)NBATCH * NTOT;
  size_t i  = (size_t)b * NTOT + p;
  out[i]       = id;                       // ids   (B,N,1)
  out[BN + i]  = sc;                       // scores(B,N,1)
  ((float4*)(out + 2 * BN))[i] = ob;       // boxes (B,N,4), 16B aligned
}

// ---------------- host launcher ----------------
extern "C" void kernel_launch(void* const* d_in, const int* in_sizes, int n_in,
                              void* d_out, int out_size, void* d_ws, size_t ws_size,
                              hipStream_t stream)
{
  (void)in_sizes; (void)n_in; (void)out_size; (void)ws_size;
  // setup_inputs() dict order: per level i: output_i, anchor_i, offset_i, stride_i
  const float* o1 = (const float*)d_in[0];
  const float* a1 = (const float*)d_in[1];
  const float* f1 = (const float*)d_in[2];
  const float* s1 = (const float*)d_in[3];
  const float* o2 = (const float*)d_in[4];
  const float* a2 = (const float*)d_in[5];
  const float* f2 = (const float*)d_in[6];
  const float* s2 = (const float*)d_in[7];
  const float* o3 = (const float*)d_in[8];
  const float* a3 = (const float*)d_in[9];
  const float* f3 = (const float*)d_in[10];
  const float* s3 = (const float*)d_in[11];

  char* ws = (char*)d_ws;
  const size_t keyBytes = (size_t)NBATCH * NTOT * sizeof(u64);      // 27.9 MB
  u64*    keysA = (u64*)ws;
  u64*    keysB = (u64*)(ws + keyBytes);
  float4* boxes = (float4*)(ws + 2 * keyBytes);                     // 18.6 MB
  u32*    cnt   = (u32*)(ws + 2 * keyBytes + (size_t)NBATCH * BOXN * sizeof(float4));
  u32*    off   = cnt + (size_t)NBATCH * TILES * RADIX;             // 3.5 MB each

  const int blkDec = (int)(((long long)NBATCH * BOXN + 255) / 256);
  const int blkOut = (int)(((long long)NBATCH * NTOT + 255) / 256);

  k_decode<<<blkDec, 256, 0, stream>>>(o1, a1, f1, s1, o2, a2, f2, s2, o3, a3, f3, s3,
                                       keysA, boxes);

  // 5 x 9-bit LSD radix passes cover the 45-bit key exactly
  u64* s = keysA; u64* d = keysB;
  for (int pass = 0; pass < 5; ++pass) {
    int shift = pass * RBITS;
    dim3 g(TILES, NBATCH);
    k_hist   <<<g, TPB, 0, stream>>>(s, cnt, shift);
    k_scan   <<<NBATCH, RADIX, 0, stream>>>(cnt, off);
    k_scatter<<<g, TPB, 0, stream>>>(s, d, off, shift);
    u64* tmp = s; s = d; d = tmp;
  }
  // after 5 passes the sorted keys are in `s` (== keysB)
  k_output<<<blkOut, 256, 0, stream>>>(s, boxes, (float*)d_out);
}